// VariationalHypergraphAutoencoder_65867618451721
// MI455X (gfx1250) — compile-verified
//
#include <hip/hip_runtime.h>

typedef __attribute__((ext_vector_type(2))) float v2f;
typedef __attribute__((ext_vector_type(4))) float v4f;
typedef __attribute__((ext_vector_type(8))) float v8f;

#define HID       128
#define N_NODES   50000
#define N_HEDGES  10000
#define NNZ       1600000
#define LDS_STRIDE 132   // padded to break 128-float stride bank conflicts

// ---------------------------------------------------------------------------
// utility: zero a float buffer
// ---------------------------------------------------------------------------
__global__ void zero_f32_kernel(float* __restrict__ p, int n) {
  int i = blockIdx.x * blockDim.x + threadIdx.x;
  if (i < n) p[i] = 0.0f;
}

// ---------------------------------------------------------------------------
// degree counts: Dn[node]++, Be[hedge]++
// ---------------------------------------------------------------------------
__global__ void degree_kernel(const int* __restrict__ node_idx,
                              const int* __restrict__ hedge_idx,
                              float* __restrict__ Dn, float* __restrict__ Be) {
  int i = blockIdx.x * blockDim.x + threadIdx.x;
  if (i < NNZ) {
    atomicAdd(&Dn[node_idx[i]], 1.0f);
    atomicAdd(&Be[hedge_idx[i]], 1.0f);
  }
}

// ---------------------------------------------------------------------------
// shared helpers for the WMMA GEMMs
// Block: 256 threads = 8 waves. Tile: 64 rows x 128 cols.
// Wave w owns column tile n0 = 16*w and 4 row tiles; B-frag reused 4x.
// A staged in LDS (padded stride); B read per-lane from global (L2-resident).
// ---------------------------------------------------------------------------
__device__ __forceinline__ void stage_A_tile(float* Alds, const float* __restrict__ X,
                                             int m_base, int M, int tid) {
#pragma unroll
  for (int t = 0; t < 8; ++t) {
    int idx = tid + t * 256;          // 2048 float4 slots = 64 rows * 32
    int r  = idx >> 5;
    int c4 = (idx & 31) << 2;
    v4f v = {0.f, 0.f, 0.f, 0.f};
    int row = m_base + r;
    if (row < M) v = *(const v4f*)(X + (long)row * HID + c4);
    *(v4f*)(&Alds[r * LDS_STRIDE + c4]) = v;
  }
}

__device__ __forceinline__ void prefetch_w_row(const float* p) {
  // 128 floats = 512B per weight row: prefetch its cachelines while A stages
  __builtin_prefetch(p +  0, 0, 0);
  __builtin_prefetch(p + 32, 0, 0);
  __builtin_prefetch(p + 64, 0, 0);
  __builtin_prefetch(p + 96, 0, 0);
}

// ---------------------------------------------------------------------------
// WMMA f32 GEMM:  Y[M,128] = X[M,128] @ W^T   (W is [128,128] row-major)
// ---------------------------------------------------------------------------
__global__ __launch_bounds__(256) void gemm_xwt_kernel(
    const float* __restrict__ X, const float* __restrict__ W,
    float* __restrict__ Y, int M) {
  __shared__ float Alds[64 * LDS_STRIDE];
  const int m_base = blockIdx.x * 64;
  const int tid  = threadIdx.x;
  const int wave = tid >> 5;
  const int lane = tid & 31;
  const int half = lane >> 4;      // 0: K pair {0,1}; 1: K pair {2,3}
  const int ln   = lane & 15;      // row (A) / col (B,C)
  const int n0   = wave * 16;
  const float* Wn = W + (long)(n0 + ln) * HID;  // B[k][n] = W[n*HID + k]

  prefetch_w_row(Wn);
  stage_A_tile(Alds, X, m_base, M, tid);
  __syncthreads();

  v8f acc[4] = {v8f{}, v8f{}, v8f{}, v8f{}};
#pragma unroll 4
  for (int k0 = 0; k0 < HID; k0 += 4) {
    const int ka = k0 + 2 * half;
    v2f b  = *(const v2f*)(Wn + ka);
    v2f a0 = *(const v2f*)(&Alds[(ln +  0) * LDS_STRIDE + ka]);
    v2f a1 = *(const v2f*)(&Alds[(ln + 16) * LDS_STRIDE + ka]);
    v2f a2 = *(const v2f*)(&Alds[(ln + 32) * LDS_STRIDE + ka]);
    v2f a3 = *(const v2f*)(&Alds[(ln + 48) * LDS_STRIDE + ka]);
    acc[0] = __builtin_amdgcn_wmma_f32_16x16x4_f32(false, a0, false, b, (short)0, acc[0], false, false);
    acc[1] = __builtin_amdgcn_wmma_f32_16x16x4_f32(false, a1, false, b, (short)0, acc[1], false, false);
    acc[2] = __builtin_amdgcn_wmma_f32_16x16x4_f32(false, a2, false, b, (short)0, acc[2], false, false);
    acc[3] = __builtin_amdgcn_wmma_f32_16x16x4_f32(false, a3, false, b, (short)0, acc[3], false, false);
  }

  const int col = n0 + ln;
  if (m_base + 64 <= M) {
    // interior block: all 64 rows valid -> unguarded stores
#pragma unroll
    for (int mt = 0; mt < 4; ++mt) {
#pragma unroll
      for (int r = 0; r < 8; ++r) {
        int row = m_base + mt * 16 + r + 8 * half;
        Y[(long)row * HID + col] = acc[mt][r];
      }
    }
  } else {
#pragma unroll
    for (int mt = 0; mt < 4; ++mt) {
#pragma unroll
      for (int r = 0; r < 8; ++r) {
        int row = m_base + mt * 16 + r + 8 * half;
        if (row < M) Y[(long)row * HID + col] = acc[mt][r];
      }
    }
  }
}

// ---------------------------------------------------------------------------
// scatter-add of 128-float rows: dst[dst_idx[i]] += src[src_idx[i]]
// One wave per nz edge: lane handles 4 columns (float4 gather + 4 atomics).
// ---------------------------------------------------------------------------
__global__ __launch_bounds__(256) void scatter_add_rows_kernel(
    const int* __restrict__ src_idx, const int* __restrict__ dst_idx,
    const float* __restrict__ src, float* __restrict__ dst, int nnz) {
  int i = blockIdx.x * 8 + (threadIdx.x >> 5);
  if (i >= nnz) return;
  int lane = threadIdx.x & 31;
  int s = src_idx[i];
  int d = dst_idx[i];
  v4f v = *(const v4f*)(src + (long)s * HID + lane * 4);
  float* o = dst + (long)d * HID + lane * 4;
  atomicAdd(o + 0, v.x);
  atomicAdd(o + 1, v.y);
  atomicAdd(o + 2, v.z);
  atomicAdd(o + 3, v.w);
}

// ---------------------------------------------------------------------------
// e[h][:] *= (Be[h] > 0 ? 1/Be[h] : 0)
// ---------------------------------------------------------------------------
__global__ void scale_e_kernel(float* __restrict__ e, const float* __restrict__ Be, int n) {
  int i = blockIdx.x * blockDim.x + threadIdx.x;
  if (i < n) {
    float d = Be[i >> 7];
    e[i] *= (d > 0.0f) ? (1.0f / d) : 0.0f;
  }
}

// ---------------------------------------------------------------------------
// h = relu(Dinv * agg + b_conv)
// ---------------------------------------------------------------------------
__global__ void finish_conv_kernel(const float* __restrict__ agg,
                                   const float* __restrict__ Dn,
                                   const float* __restrict__ b,
                                   float* __restrict__ h, int n) {
  int i = blockIdx.x * blockDim.x + threadIdx.x;
  if (i < n) {
    float d = Dn[i >> 7];
    float dinv = (d > 0.0f) ? (1.0f / d) : 0.0f;
    float v = dinv * agg[i] + b[i & (HID - 1)];
    h[i] = fmaxf(v, 0.0f);
  }
}

// ---------------------------------------------------------------------------
// Fused VAE head: per 16x16 tile compute mean = H@Wm^T+bm and ls = H@Ws^T+bs
// in registers (shared A-frags, 8 WMMAs per k-step), then z = noise*exp(ls)+mean.
// ---------------------------------------------------------------------------
__global__ __launch_bounds__(256) void vae_head_kernel(
    const float* __restrict__ H,  const float* __restrict__ Wm,
    const float* __restrict__ Ws, const float* __restrict__ bm,
    const float* __restrict__ bs, const float* __restrict__ noise,
    float* __restrict__ Z, int M) {
  __shared__ float Alds[64 * LDS_STRIDE];
  const int m_base = blockIdx.x * 64;
  const int tid  = threadIdx.x;
  const int wave = tid >> 5;
  const int lane = tid & 31;
  const int half = lane >> 4;
  const int ln   = lane & 15;
  const int n0   = wave * 16;
  const float* Wmn = Wm + (long)(n0 + ln) * HID;
  const float* Wsn = Ws + (long)(n0 + ln) * HID;

  prefetch_w_row(Wmn);
  prefetch_w_row(Wsn);
  stage_A_tile(Alds, H, m_base, M, tid);
  __syncthreads();

  v8f am[4] = {v8f{}, v8f{}, v8f{}, v8f{}};
  v8f as[4] = {v8f{}, v8f{}, v8f{}, v8f{}};
#pragma unroll 2
  for (int k0 = 0; k0 < HID; k0 += 4) {
    const int ka = k0 + 2 * half;
    v2f bmv = *(const v2f*)(Wmn + ka);
    v2f bsv = *(const v2f*)(Wsn + ka);
    v2f a0 = *(const v2f*)(&Alds[(ln +  0) * LDS_STRIDE + ka]);
    v2f a1 = *(const v2f*)(&Alds[(ln + 16) * LDS_STRIDE + ka]);
    v2f a2 = *(const v2f*)(&Alds[(ln + 32) * LDS_STRIDE + ka]);
    v2f a3 = *(const v2f*)(&Alds[(ln + 48) * LDS_STRIDE + ka]);
    am[0] = __builtin_amdgcn_wmma_f32_16x16x4_f32(false, a0, false, bmv, (short)0, am[0], false, false);
    am[1] = __builtin_amdgcn_wmma_f32_16x16x4_f32(false, a1, false, bmv, (short)0, am[1], false, false);
    am[2] = __builtin_amdgcn_wmma_f32_16x16x4_f32(false, a2, false, bmv, (short)0, am[2], false, false);
    am[3] = __builtin_amdgcn_wmma_f32_16x16x4_f32(false, a3, false, bmv, (short)0, am[3], false, false);
    as[0] = __builtin_amdgcn_wmma_f32_16x16x4_f32(false, a0, false, bsv, (short)0, as[0], false, false);
    as[1] = __builtin_amdgcn_wmma_f32_16x16x4_f32(false, a1, false, bsv, (short)0, as[1], false, false);
    as[2] = __builtin_amdgcn_wmma_f32_16x16x4_f32(false, a2, false, bsv, (short)0, as[2], false, false);
    as[3] = __builtin_amdgcn_wmma_f32_16x16x4_f32(false, a3, false, bsv, (short)0, as[3], false, false);
  }

  const int col = n0 + ln;
  const float biasm = bm[col];
  const float biass = bs[col];
  if (m_base + 64 <= M) {
    // interior block: all 64 rows valid -> unguarded epilogue
#pragma unroll
    for (int mt = 0; mt < 4; ++mt) {
#pragma unroll
      for (int r = 0; r < 8; ++r) {
        int row = m_base + mt * 16 + r + 8 * half;
        long off = (long)row * HID + col;
        float mean = am[mt][r] + biasm;
        float ls   = as[mt][r] + biass;
        Z[off] = noise[off] * expf(ls) + mean;
      }
    }
  } else {
#pragma unroll
    for (int mt = 0; mt < 4; ++mt) {
#pragma unroll
      for (int r = 0; r < 8; ++r) {
        int row = m_base + mt * 16 + r + 8 * half;
        if (row < M) {
          long off = (long)row * HID + col;
          float mean = am[mt][r] + biasm;
          float ls   = as[mt][r] + biass;
          Z[off] = noise[off] * expf(ls) + mean;
        }
      }
    }
  }
}

// ---------------------------------------------------------------------------
// launch
// ---------------------------------------------------------------------------
extern "C" void kernel_launch(void* const* d_in, const int* in_sizes, int n_in,
                              void* d_out, int out_size, void* d_ws, size_t ws_size,
                              hipStream_t stream) {
  (void)in_sizes; (void)n_in; (void)out_size; (void)ws_size;
  const float* x     = (const float*)d_in[0];
  const int*   nidx  = (const int*)d_in[1];
  const int*   hidx  = (const int*)d_in[2];
  const float* noise = (const float*)d_in[3];
  const float* Wc    = (const float*)d_in[4];
  const float* bc    = (const float*)d_in[5];
  const float* Wm    = (const float*)d_in[6];
  const float* bm    = (const float*)d_in[7];
  const float* Wsd   = (const float*)d_in[8];
  const float* bsd   = (const float*)d_in[9];
  float* Z  = (float*)d_out;
  float* ws = (float*)d_ws;

  // workspace layout (floats); agg aliases xt (xt fully consumed before agg init)
  float* Dn  = ws;                               // N_NODES
  float* Be  = Dn + N_NODES;                     // N_HEDGES
  float* xt  = Be + N_HEDGES;                    // N_NODES*HID
  float* agg = xt;                               // alias (stream-ordered safe)
  float* e   = xt + (size_t)N_NODES * HID;       // N_HEDGES*HID
  float* h   = e  + (size_t)N_HEDGES * HID;      // N_NODES*HID

  {
    int n = N_NODES + N_HEDGES;                  // Dn & Be contiguous
    zero_f32_kernel<<<(n + 255) / 256, 256, 0, stream>>>(Dn, n);
  }
  {
    int n = N_HEDGES * HID;
    zero_f32_kernel<<<(n + 255) / 256, 256, 0, stream>>>(e, n);
  }

  degree_kernel<<<(NNZ + 255) / 256, 256, 0, stream>>>(nidx, hidx, Dn, Be);

  gemm_xwt_kernel<<<(N_NODES + 63) / 64, 256, 0, stream>>>(x, Wc, xt, N_NODES);

  // node -> hyperedge: e[hedge] += xt[node]
  scatter_add_rows_kernel<<<(NNZ + 7) / 8, 256, 0, stream>>>(nidx, hidx, xt, e, NNZ);

  {
    int n = N_HEDGES * HID;
    scale_e_kernel<<<(n + 255) / 256, 256, 0, stream>>>(e, Be, n);
  }
  {
    int n = N_NODES * HID;
    zero_f32_kernel<<<(n + 255) / 256, 256, 0, stream>>>(agg, n);
  }

  // hyperedge -> node: agg[node] += e[hedge]
  scatter_add_rows_kernel<<<(NNZ + 7) / 8, 256, 0, stream>>>(hidx, nidx, e, agg, NNZ);

  {
    int n = N_NODES * HID;
    finish_conv_kernel<<<(n + 255) / 256, 256, 0, stream>>>(agg, Dn, bc, h, n);
  }

  vae_head_kernel<<<(N_NODES + 63) / 64, 256, 0, stream>>>(h, Wm, Wsd, bm, bsd, noise, Z, N_NODES);
}